// HAST_11149735100759
// MI455X (gfx1250) — compile-verified
//
#include <hip/hip_runtime.h>
#include <hip/hip_bf16.h>
#include <math.h>

typedef __attribute__((ext_vector_type(2))) float v2f;
typedef __attribute__((ext_vector_type(8))) float v8f;

#if defined(__AMDGCN__) && __has_builtin(__builtin_amdgcn_wmma_f32_16x16x4_f32)
#define USE_WMMA_F32 1
#else
#define USE_WMMA_F32 0
#endif

// D(16x16) += A(16x4) x B(4x16), fp32, wave32. Fragment layout per ISA 7.12.2.
__device__ __forceinline__ v8f wmma4v(v2f a, v2f b, v8f c) {
#if USE_WMMA_F32
  return __builtin_amdgcn_wmma_f32_16x16x4_f32(false, a, false, b, (short)0, c, false, false);
#else
  (void)a; (void)b;
  return c;
#endif
}

__device__ __forceinline__ float sigf(float x) { return 1.0f / (1.0f + expf(-x)); }

// ---------------------------------------------------------------------------
// WMMA GEMM, one wave per 16x64 C strip (4 accumulators, A reused 4x).
// Contract (all launches comply): K % 4 == 0, M % 16 == 0, A 8B-aligned with
// even lda; transB requires even ldb. N edge handled by clamped loads +
// guarded stores. transB: B(k,n) at Bm[n*ldb+k]; else Bm[k*ldb+n].
// Grid: (ceil(N/64), M/16, batch); block = 32.
// ---------------------------------------------------------------------------
__global__ void gemm_wmma_f32(const float* __restrict__ A, int lda, long long sA,
                              const float* __restrict__ Bm, int ldb, long long sB,
                              int transB,
                              float* __restrict__ C, int ldc, long long sC,
                              int M, int N, int K,
                              const float* __restrict__ bias0,
                              const float* __restrict__ bias1) {
  const long long bz = blockIdx.z;
  A  += bz * sA;
  Bm += bz * sB;
  C  += bz * sC;
  const int m0 = blockIdx.y * 16;
  const int n0 = blockIdx.x * 64;
  const int lane = threadIdx.x & 31;
  const int hf = lane >> 4;       // 0: lanes 0-15 (K0,K1 / M0-7), 1: lanes 16-31
  const int lh = lane & 15;
  const int nmax = N - 1;
  const int nb0 = n0 + lh, nb1 = nb0 + 16, nb2 = nb0 + 32, nb3 = nb0 + 48;
  const int nc0 = nb0 < nmax ? nb0 : nmax;
  const int nc1 = nb1 < nmax ? nb1 : nmax;
  const int nc2 = nb2 < nmax ? nb2 : nmax;
  const int nc3 = nb3 < nmax ? nb3 : nmax;

  v8f acc0 = {0.f,0.f,0.f,0.f,0.f,0.f,0.f,0.f};
  v8f acc1 = acc0, acc2 = acc0, acc3 = acc0;

  const float* Ap = A + (long long)(m0 + lh) * lda + 2 * hf;
  const int kT = K >> 2;

  if (transB) {
    const float* B0 = Bm + (long long)nc0 * ldb + 2 * hf;
    const float* B1 = Bm + (long long)nc1 * ldb + 2 * hf;
    const float* B2 = Bm + (long long)nc2 * ldb + 2 * hf;
    const float* B3 = Bm + (long long)nc3 * ldb + 2 * hf;
#pragma unroll 2
    for (int i = 0; i < kT; ++i) {
      const v2f av = *(const v2f*)Ap;
      const v2f b0 = *(const v2f*)B0;
      const v2f b1 = *(const v2f*)B1;
      const v2f b2 = *(const v2f*)B2;
      const v2f b3 = *(const v2f*)B3;
      acc0 = wmma4v(av, b0, acc0);
      acc1 = wmma4v(av, b1, acc1);
      acc2 = wmma4v(av, b2, acc2);
      acc3 = wmma4v(av, b3, acc3);
      Ap += 4; B0 += 4; B1 += 4; B2 += 4; B3 += 4;
    }
  } else {
#pragma unroll 2
    for (int i = 0; i < kT; ++i) {
      const int k = i * 4 + 2 * hf;
      const v2f av = *(const v2f*)Ap;
      Ap += 4;
      const float* Br0 = Bm + (long long)k * ldb;
      const float* Br1 = Bm + (long long)(k + 1) * ldb;
      v2f b0, b1, b2, b3;
      b0[0] = Br0[nc0]; b0[1] = Br1[nc0];
      b1[0] = Br0[nc1]; b1[1] = Br1[nc1];
      b2[0] = Br0[nc2]; b2[1] = Br1[nc2];
      b3[0] = Br0[nc3]; b3[1] = Br1[nc3];
      acc0 = wmma4v(av, b0, acc0);
      acc1 = wmma4v(av, b1, acc1);
      acc2 = wmma4v(av, b2, acc2);
      acc3 = wmma4v(av, b3, acc3);
    }
  }

  float bv0 = 0.f, bv1 = 0.f, bv2 = 0.f, bv3 = 0.f;
  if (bias0) { bv0 = bias0[nc0]; bv1 = bias0[nc1]; bv2 = bias0[nc2]; bv3 = bias0[nc3]; }
  if (bias1) { bv0 += bias1[nc0]; bv1 += bias1[nc1]; bv2 += bias1[nc2]; bv3 += bias1[nc3]; }
#pragma unroll
  for (int r = 0; r < 8; ++r) {
    const long long m = m0 + hf * 8 + r;   // M % 16 == 0 -> always valid
    float* Crow = C + m * ldc;
    if (nb0 < N) Crow[nb0] = acc0[r] + bv0;
    if (nb1 < N) Crow[nb1] = acc1[r] + bv1;
    if (nb2 < N) Crow[nb2] = acc2[r] + bv2;
    if (nb3 < N) Crow[nb3] = acc3[r] + bv3;
  }
}

// ---------------------------------------------------------------------------
// Embedding gather: X[b*128+l][e] = table[idx[b*128+l]][e]
// ---------------------------------------------------------------------------
__global__ void embed_kernel(const int* __restrict__ idx,
                             const float* __restrict__ table,
                             float* __restrict__ X) {
  const int i = blockIdx.x * blockDim.x + threadIdx.x;
  if (i >= 2048 * 300) return;
  const int m = i / 300;
  const int e = i - m * 300;
  X[i] = table[(long long)idx[m] * 300 + e];
}

// ---------------------------------------------------------------------------
// Transpose Whh [2,400,100] (n-major) -> WhhT [4][100][400] (k-major)
// ---------------------------------------------------------------------------
__global__ void whh_transpose(const float* __restrict__ ap_Whh,
                              const float* __restrict__ op_Whh,
                              float* __restrict__ WhhT) {
  const int i = blockIdx.x * blockDim.x + threadIdx.x;
  if (i >= 4 * 40000) return;
  const int g = i / 40000;
  const int r = i - g * 40000;
  const int n = r / 100;
  const int k = r - n * 100;
  const float* src = (g >> 1) ? op_Whh : ap_Whh;
  const int dir = g & 1;
  WhhT[(long long)g * 40000 + k * 400 + n] = src[(long long)dir * 40000 + n * 100 + k];
}

// ---------------------------------------------------------------------------
// Pack biaf_W [800,401] -> Wpk [800,400] (even ld, vectorizable) + bias [800]
// ---------------------------------------------------------------------------
__global__ void pack_biaf(const float* __restrict__ biaf_W,
                          float* __restrict__ Wpk,
                          float* __restrict__ bpk) {
  const int i = blockIdx.x * blockDim.x + threadIdx.x;
  if (i < 800 * 400) {
    const int n = i / 400;
    const int k = i - n * 400;
    Wpk[i] = biaf_W[(long long)n * 401 + k];
  }
  if (i < 800) bpk[i] = biaf_W[(long long)i * 401 + 400];
}

// ---------------------------------------------------------------------------
// BiLSTM recurrence. One workgroup per (lstm,dir), 800 threads = 25 waves.
// The Whh^T strip for each wave's 16 output columns is time-invariant ->
// hoisted into 25 v2f registers. Inner loop: 25 x (ds_load_b64 + v_wmma).
// ---------------------------------------------------------------------------
__global__ void __launch_bounds__(800) lstm_rec_kernel(
    const float* __restrict__ Gin_base,   // [4][2048][400] (apF, apB, opF, opB)
    const float* __restrict__ WhhT,       // [4][100][400]
    float* __restrict__ H_ap,             // [2048][200]
    float* __restrict__ H_op) {           // [2048][200]
  __shared__ __attribute__((aligned(16))) float hbuf[16 * 100];
  __shared__ __attribute__((aligned(16))) float cbuf[16 * 100];
  __shared__ __attribute__((aligned(16))) float gates[16 * 400];
  const int g = blockIdx.x;
  const int lstm = g >> 1;
  const int dir = g & 1;
  const float* Gin = Gin_base + (long long)g * 2048 * 400;
  const float* Wt = WhhT + (long long)g * 40000;
  float* Hout = lstm ? H_op : H_ap;
  const int tid = threadIdx.x;
  for (int e = tid; e < 1600; e += 800) { hbuf[e] = 0.f; cbuf[e] = 0.f; }
  const int wave = tid >> 5;
  const int lane = tid & 31;
  const int hf = lane >> 4;
  const int lh = lane & 15;
  const int n0 = wave * 16;

  // Hoist B fragments: time-invariant Whh^T columns for this wave.
  v2f breg[25];
#pragma unroll
  for (int i = 0; i < 25; ++i) {
    const int k = i * 4 + 2 * hf;
    breg[i][0] = Wt[k * 400 + n0 + lh];
    breg[i][1] = Wt[(k + 1) * 400 + n0 + lh];
  }
  __syncthreads();

  const v2f* hrow = (const v2f*)(hbuf + lh * 100);   // lh*100 even -> 8B aligned
  const int n = n0 + lh;
  for (int t = 0; t < 128; ++t) {
    const int teff = dir ? (127 - t) : t;
    v8f acc = {0.f,0.f,0.f,0.f,0.f,0.f,0.f,0.f};
#pragma unroll
    for (int i = 0; i < 25; ++i) {
      const v2f av = hrow[i * 2 + hf];               // h[lh][i*4 + 2*hf .. +1]
      acc = wmma4v(av, breg[i], acc);
    }
#pragma unroll
    for (int r = 0; r < 8; ++r) {
      const int m = hf * 8 + r;                      // batch row
      gates[m * 400 + n] = acc[r] + Gin[((long long)m * 128 + teff) * 400 + n];
    }
    __syncthreads();
    for (int e = tid; e < 1600; e += 800) {
      const int b = e / 100;
      const int j = e - b * 100;
      const float gi = gates[b * 400 + j];
      const float gf = gates[b * 400 + 100 + j];
      const float gg = gates[b * 400 + 200 + j];
      const float go = gates[b * 400 + 300 + j];
      const float c0 = cbuf[e];
      const float cn = sigf(gf) * c0 + sigf(gi) * tanhf(gg);
      const float hn = sigf(go) * tanhf(cn);
      cbuf[e] = cn;
      hbuf[e] = hn;
      Hout[((long long)b * 128 + teff) * 200 + dir * 100 + j] = hn;
    }
    __syncthreads();
  }
}

// ---------------------------------------------------------------------------
// Truncated-history attention. One block per batch element (16 blocks).
// Rolling caches P1 = Hp@W1 rows, P3 = Hpt@W3 rows -> 3 length-200 GEMVs/step.
// Writes ht_tilde into ap_feat[:, 0:200] (ld 400).
// ---------------------------------------------------------------------------
__global__ void tha_kernel(const float* __restrict__ H_ap,
                           const float* __restrict__ W1,
                           const float* __restrict__ W2,
                           const float* __restrict__ W3,
                           const float* __restrict__ vvec,
                           float* __restrict__ ap_feat) {
  __shared__ float P1[5][200];
  __shared__ float P3[5][200];
  __shared__ float Hpt[5][200];
  __shared__ float ht[200];
  __shared__ float htt[200];
  __shared__ float red[5][200];
  __shared__ float wj[5];
  const int b = blockIdx.x;
  const int tid = threadIdx.x;
  for (int e = tid; e < 5 * 200; e += 256) {
    ((float*)P1)[e] = 0.f; ((float*)P3)[e] = 0.f; ((float*)Hpt)[e] = 0.f;
  }
  __syncthreads();
  for (int t = 0; t < 128; ++t) {
    if (tid < 200) ht[tid] = H_ap[((long long)b * 128 + t) * 200 + tid];
    __syncthreads();
    if (tid < 200) {
      const int d = tid;
      float q = 0.f;
      for (int k = 0; k < 200; ++k) q += ht[k] * W2[k * 200 + d];
      const float vd = vvec[d];
#pragma unroll
      for (int j = 0; j < 5; ++j)
        red[j][d] = tanhf(P1[j][d] + q + P3[j][d]) * vd;
    }
    __syncthreads();
    if (tid < 5) {
      float s = 0.f;
      for (int d = 0; d < 200; ++d) s += red[tid][d];
      wj[tid] = s;
    }
    __syncthreads();
    if (tid == 0) {
      float mx = wj[0];
#pragma unroll
      for (int j = 1; j < 5; ++j) mx = fmaxf(mx, wj[j]);
      float sum = 0.f, e_[5];
#pragma unroll
      for (int j = 0; j < 5; ++j) { e_[j] = expf(wj[j] - mx); sum += e_[j]; }
#pragma unroll
      for (int j = 0; j < 5; ++j) wj[j] = e_[j] / sum;
    }
    __syncthreads();
    if (tid < 200) {
      const int d = tid;
      float hh = 0.f;
#pragma unroll
      for (int j = 0; j < 5; ++j) hh += wj[j] * Hpt[j][d];
      const float out = ht[d] + fmaxf(hh, 0.f);
      htt[d] = out;
      ap_feat[((long long)b * 128 + t) * 400 + d] = out;
    }
    __syncthreads();
    if (tid < 200) {
      const int d = tid;
#pragma unroll
      for (int j = 0; j < 4; ++j) {
        P1[j][d] = P1[j + 1][d];
        P3[j][d] = P3[j + 1][d];
        Hpt[j][d] = Hpt[j + 1][d];
      }
      float p1 = 0.f, p3 = 0.f;
      for (int k = 0; k < 200; ++k) {
        p1 += ht[k] * W1[k * 200 + d];
        p3 += htt[k] * W3[k * 200 + d];
      }
      P1[4][d] = p1;
      P3[4][d] = p3;
      Hpt[4][d] = htt[d];
    }
    __syncthreads();
  }
}

// ---------------------------------------------------------------------------
// STN core, fused per (b,t). Writes summary into ap_feat[:, 200:400].
// ---------------------------------------------------------------------------
__global__ void stn_kernel(const float* __restrict__ A_stn,
                           const float* __restrict__ O_stn,
                           const float* __restrict__ bil,
                           const float* __restrict__ H_op,
                           const float* __restrict__ stn_b,
                           const float* __restrict__ stn_b_bil,
                           float* __restrict__ ap_feat) {
  __shared__ float At[200];
  __shared__ float Bt[200];
  __shared__ float Ws[128];
  const int bt = blockIdx.x;
  const int b = bt >> 7;
  const int tid = threadIdx.x;
  if (tid < 200) {
    At[tid] = A_stn[(long long)bt * 200 + tid] + stn_b[tid];
    Bt[tid] = bil[(long long)bt * 200 + tid];
  }
  __syncthreads();
  if (tid < 128) {
    const int s = tid;
    const float* HOrow = H_op + ((long long)b * 128 + s) * 200;
    const float* Orow = O_stn + ((long long)b * 128 + s) * 200;
    float acc = 0.f;
    for (int d = 0; d < 200; ++d) {
      const float hh = HOrow[d] + fmaxf(At[d] + Orow[d], 0.f);
      acc += hh * Bt[d];
    }
    Ws[s] = tanhf(acc + stn_b_bil[0]);
  }
  __syncthreads();
  if (tid == 0) {
    float mx = Ws[0];
    for (int s = 1; s < 128; ++s) mx = fmaxf(mx, Ws[s]);
    float sum = 0.f;
    for (int s = 0; s < 128; ++s) { Ws[s] = expf(Ws[s] - mx); sum += Ws[s]; }
    const float inv = 1.f / sum;
    for (int s = 0; s < 128; ++s) Ws[s] *= inv;
  }
  __syncthreads();
  if (tid < 200) {
    const int d = tid;
    const float a = At[d];
    float acc = 0.f;
    for (int s = 0; s < 128; ++s) {
      const long long row = (long long)b * 128 + s;
      const float hh = H_op[row * 200 + d] + fmaxf(a + O_stn[row * 200 + d], 0.f);
      acc += Ws[s] * hh;
    }
    ap_feat[(long long)bt * 400 + 200 + d] = acc;
  }
}

// ---------------------------------------------------------------------------
extern "C" void kernel_launch(void* const* d_in, const int* in_sizes, int n_in,
                              void* d_out, int out_size, void* d_ws, size_t ws_size,
                              hipStream_t stream) {
  (void)in_sizes; (void)n_in; (void)out_size; (void)ws_size;
  const int* text_indices = (const int*)d_in[0];
  // d_in[1] = text_mask (all ones -> plain BiLSTM, unused)
  const float* embed_table = (const float*)d_in[2];
  const float* ap_Wih = (const float*)d_in[3];
  const float* ap_Whh = (const float*)d_in[4];
  const float* ap_bih = (const float*)d_in[5];
  const float* ap_bhh = (const float*)d_in[6];
  const float* op_Wih = (const float*)d_in[7];
  const float* op_Whh = (const float*)d_in[8];
  const float* op_bih = (const float*)d_in[9];
  const float* op_bhh = (const float*)d_in[10];
  const float* tha_v  = (const float*)d_in[11];
  const float* tha_W1 = (const float*)d_in[12];
  const float* tha_W2 = (const float*)d_in[13];
  const float* tha_W3 = (const float*)d_in[14];
  const float* stn_W_A = (const float*)d_in[15];
  const float* stn_W_O = (const float*)d_in[16];
  const float* stn_b   = (const float*)d_in[17];
  const float* stn_W_bil = (const float*)d_in[18];
  const float* stn_b_bil = (const float*)d_in[19];
  const float* biaf_W  = (const float*)d_in[20];
  const float* ap_fc_W = (const float*)d_in[21];
  const float* ap_fc_b = (const float*)d_in[22];
  const float* op_fc_W = (const float*)d_in[23];
  const float* op_fc_b = (const float*)d_in[24];

  float* out = (float*)d_out;     // [ap_out 10240 | op_out 10240 | triplet 1048576]

  // Workspace layout (floats). Total ~8.88M floats = ~35.5 MB.
  float* ws = (float*)d_ws;
  float* X       = ws;                      // 2048*300   = 614400
  float* Gin     = X + 614400;              // 4*2048*400 = 3276800
  float* WhhT    = Gin + 3276800;           // 4*100*400  = 160000
  float* H_ap    = WhhT + 160000;           // 2048*200   = 409600
  float* H_op    = H_ap + 409600;           // 409600
  float* ap_feat = H_op + 409600;           // 2048*400   = 819200
  float* A_stn   = ap_feat + 819200;        // 409600
  float* O_stn   = A_stn + 409600;          // 409600
  float* bilbuf  = O_stn + 409600;          // 409600
  float* affine  = bilbuf + 409600;         // 2048*800   = 1638400
  float* Wpk     = affine + 1638400;        // 800*400    = 320000
  float* bpk     = Wpk + 320000;            // 800

  // 1) Embedding gather
  embed_kernel<<<(2048 * 300 + 255) / 256, 256, 0, stream>>>(text_indices, embed_table, X);

  // 2) Input projections + gate biases: Gin[g] = X @ Wih[dir].T + bih + bhh
  for (int g = 0; g < 4; ++g) {
    const int lstm = g >> 1, dir = g & 1;
    const float* Wih = (lstm ? op_Wih : ap_Wih) + (long long)dir * 400 * 300;
    const float* bih = (lstm ? op_bih : ap_bih) + dir * 400;
    const float* bhh = (lstm ? op_bhh : ap_bhh) + dir * 400;
    gemm_wmma_f32<<<dim3(7, 128, 1), 32, 0, stream>>>(
        X, 300, 0, Wih, 300, 0, /*transB=*/1,
        Gin + (long long)g * 819200, 400, 0,
        2048, 400, 300, bih, bhh);
  }

  // 3) Pre-transpose Whh; pack biaffine weight to even ld
  whh_transpose<<<(160000 + 255) / 256, 256, 0, stream>>>(ap_Whh, op_Whh, WhhT);
  pack_biaf<<<(800 * 400 + 255) / 256, 256, 0, stream>>>(biaf_W, Wpk, bpk);

  // 4) Four BiLSTM directions concurrently (4 workgroups)
  lstm_rec_kernel<<<4, 800, 0, stream>>>(Gin, WhhT, H_ap, H_op);

  // 5) Truncated-history attention -> ap_feat[:, 0:200]
  tha_kernel<<<16, 256, 0, stream>>>(H_ap, tha_W1, tha_W2, tha_W3, tha_v, ap_feat);

  // 6) STN projections
  gemm_wmma_f32<<<dim3(4, 128, 1), 32, 0, stream>>>(
      ap_feat, 400, 0, stn_W_A, 200, 0, 0, A_stn, 200, 0, 2048, 200, 200, nullptr, nullptr);
  gemm_wmma_f32<<<dim3(4, 128, 1), 32, 0, stream>>>(
      H_op, 200, 0, stn_W_O, 200, 0, 0, O_stn, 200, 0, 2048, 200, 200, nullptr, nullptr);
  gemm_wmma_f32<<<dim3(4, 128, 1), 32, 0, stream>>>(
      ap_feat, 400, 0, stn_W_bil, 200, 0, 0, bilbuf, 200, 0, 2048, 200, 200, nullptr, nullptr);

  // 7) Fused STN -> ap_feat[:, 200:400]
  stn_kernel<<<2048, 256, 0, stream>>>(A_stn, O_stn, bilbuf, H_op, stn_b, stn_b_bil, ap_feat);

  // 8) Biaffine affine: ap_feat[2048,400] @ Wpk.T + bpk
  gemm_wmma_f32<<<dim3(13, 128, 1), 32, 0, stream>>>(
      ap_feat, 400, 0, Wpk, 400, 0, 1, affine, 800, 0,
      2048, 800, 400, bpk, nullptr);

  // 9) FC heads -> d_out
  gemm_wmma_f32<<<dim3(1, 128, 1), 32, 0, stream>>>(
      ap_feat, 400, 0, ap_fc_W, 400, 0, 1, out, 5, 0, 2048, 5, 400, ap_fc_b, nullptr);
  gemm_wmma_f32<<<dim3(1, 128, 1), 32, 0, stream>>>(
      H_op, 200, 0, op_fc_W, 200, 0, 1, out + 10240, 5, 0, 2048, 5, 200, op_fc_b, nullptr);

  // 10) Biaffine einsum as batched GEMM:
  //     triplet[b][y][x*4+o] = sum_d H_op[b,y,d] * affine[b, x*4+o, d]
  gemm_wmma_f32<<<dim3(8, 8, 16), 32, 0, stream>>>(
      H_op, 200, 25600, affine, 200, 102400, 1,
      out + 20480, 512, 65536, 128, 512, 200, nullptr, nullptr);
}